// PolyMlp_SkipLoRA_47536698032189
// MI455X (gfx1250) — compile-verified
//
#include <hip/hip_runtime.h>
#include <stdint.h>

typedef __attribute__((ext_vector_type(2))) float v2f;
typedef __attribute__((ext_vector_type(8))) float v8f;

#define TM 128   // block tile rows
#define TN 64    // block tile cols
#define TK 32    // k step
#define LDSP 40  // padded LDS row stride (floats), even & 16B-friendly

// D[M,N] = A[M,K] . W[N,K]^T  (+ bias[N]) (+ D if accum)
// Requires M%128==0, N%64==0, K%32==0 (all shapes here satisfy this).
// Double-buffered LDS fed by CDNA5 async global->LDS copies (ASYNCcnt).
__global__ __launch_bounds__(256)
void gemm_f32_wmma(const float* __restrict__ A, const float* __restrict__ W,
                   const float* __restrict__ bias, float* __restrict__ D,
                   int M, int N, int K, int accum)
{
    __shared__ __align__(16) float As[2][TM][LDSP];
    __shared__ __align__(16) float Ws[2][TN][LDSP];

    const int tid   = threadIdx.x;
    const int lane  = tid & 31;
    const int wave  = tid >> 5;           // 0..7 -> 16-row strip
    const int ln15  = lane & 15;
    const int khalf = (lane >> 4) << 1;   // 0 or 2 (fp32 WMMA A/B lane split)

    const int rowBase = blockIdx.y * TM;
    const int colBase = blockIdx.x * TN;

    // LDS byte offsets: flat address of a __shared__ object keeps the LDS
    // offset in its low 32 bits (aperture lives in addr[63:32]).
    const uint32_t asB = (uint32_t)(uintptr_t)(void*)&As[0][0][0];
    const uint32_t wsB = (uint32_t)(uintptr_t)(void*)&Ws[0][0][0];
    const uint32_t abuf = (uint32_t)(TM * LDSP * 4);
    const uint32_t wbuf = (uint32_t)(TN * LDSP * 4);

    // Issue one tile's async global->LDS copies (6 x b128 per thread).
    auto stage = [&](int k0, int buf) {
#pragma unroll
        for (int i = 0; i < 4; ++i) {               // A: 128x32 = 1024 float4
            int f  = tid + i * 256;
            int r  = f >> 3;
            int kq = (f & 7) << 2;
            const float* g = A + (size_t)(rowBase + r) * K + k0 + kq;
            uint32_t l = asB + (uint32_t)buf * abuf + (uint32_t)(r * LDSP + kq) * 4u;
            asm volatile("global_load_async_to_lds_b128 %0, %1, off"
                         :: "v"(l), "v"(g) : "memory");
        }
#pragma unroll
        for (int i = 0; i < 2; ++i) {               // W: 64x32 = 512 float4
            int f  = tid + i * 256;
            int r  = f >> 3;
            int kq = (f & 7) << 2;
            const float* g = W + (size_t)(colBase + r) * K + k0 + kq;
            uint32_t l = wsB + (uint32_t)buf * wbuf + (uint32_t)(r * LDSP + kq) * 4u;
            asm volatile("global_load_async_to_lds_b128 %0, %1, off"
                         :: "v"(l), "v"(g) : "memory");
        }
    };

    v8f acc[4];
#pragma unroll
    for (int j = 0; j < 4; ++j) acc[j] = (v8f){0.f,0.f,0.f,0.f,0.f,0.f,0.f,0.f};

    const int nk = K / TK;
    stage(0, 0);                                    // prefetch tile 0

    for (int kt = 0; kt < nk; ++kt) {
        const int buf = kt & 1;
        // Own async copies for tile kt have landed in LDS:
        asm volatile("s_wait_asynccnt 0x0" ::: "memory");
        // Everyone's copies landed AND everyone finished compute on buf^1:
        __syncthreads();
        // Prefetch next tile into the now-idle buffer, overlapped with WMMAs.
        if (kt + 1 < nk) stage((kt + 1) * TK, buf ^ 1);

        const int arow = wave * 16 + ln15;
#pragma unroll
        for (int ks = 0; ks < TK; ks += 4) {
            const int kk = ks + khalf;
            v2f a = *(const v2f*)&As[buf][arow][kk];
#pragma unroll
            for (int j = 0; j < 4; ++j) {
                v2f b = *(const v2f*)&Ws[buf][j * 16 + ln15][kk];
                acc[j] = __builtin_amdgcn_wmma_f32_16x16x4_f32(
                    false, a, false, b, (short)0, acc[j], false, false);
            }
        }
    }

    // Epilogue: C/D layout -> lane<16: M=e, lane>=16: M=e+8; N = ln15 (+16j)
    const int rlo = rowBase + wave * 16 + ((lane >> 4) << 3);
#pragma unroll
    for (int j = 0; j < 4; ++j) {
        const int c   = colBase + j * 16 + ln15;
        const float bv = bias ? bias[c] : 0.0f;
#pragma unroll
        for (int e = 0; e < 8; ++e) {
            const size_t idx = (size_t)(rlo + e) * N + c;
            float val = acc[j][e] + bv;
            if (accum) val += D[idx];
            D[idx] = val;
        }
    }
}

// Per-row LayerNorm of out1 & out3 + alpha-gated product, in-place over out1.
__global__ __launch_bounds__(256)
void ln_gate_kernel(float* __restrict__ o1, const float* __restrict__ out3,
                    const float* __restrict__ g1, const float* __restrict__ b1,
                    const float* __restrict__ g3, const float* __restrict__ b3,
                    const float* __restrict__ alpha_p, int M, int Dd)
{
    const int row  = blockIdx.x * 8 + (threadIdx.x >> 5);
    const int lane = threadIdx.x & 31;
    if (row >= M) return;

    const float* r1 = o1   + (size_t)row * Dd;
    const float* r3 = out3 + (size_t)row * Dd;

    float s1 = 0.f, q1 = 0.f, s3 = 0.f, q3 = 0.f;
    for (int c = lane; c < Dd; c += 32) {
        float v1 = r1[c], v3 = r3[c];
        s1 += v1; q1 += v1 * v1;
        s3 += v3; q3 += v3 * v3;
    }
#pragma unroll
    for (int off = 16; off > 0; off >>= 1) {
        s1 += __shfl_xor(s1, off, 32);
        q1 += __shfl_xor(q1, off, 32);
        s3 += __shfl_xor(s3, off, 32);
        q3 += __shfl_xor(q3, off, 32);
    }
    const float inv = 1.0f / (float)Dd;
    const float mu1 = s1 * inv, mu3 = s3 * inv;
    const float rs1 = rsqrtf(q1 * inv - mu1 * mu1 + 1e-6f);
    const float rs3 = rsqrtf(q3 * inv - mu3 * mu3 + 1e-6f);
    const float al  = alpha_p[0];

    float* w1 = o1 + (size_t)row * Dd;
    for (int c = lane; c < Dd; c += 32) {
        float v1 = (r1[c] - mu1) * rs1 * g1[c] + b1[c];
        float v3 = (r3[c] - mu3) * rs3 * g3[c] + b3[c];
        w1[c] = v1 + al * (v1 * v3);
    }
}

// Pack LoRA factors into padded GEMM operands.
// A_pack [128][1024]:  row j=(t*R+r) = lora_A[t][r][:]  (lora_A contiguous in j)
// B_pack [1024][128]:  B_pack[o][j]  = lora_B[t][o][r]
// j >= (task+1)*R -> zero; task < 0 -> all zero (LoRA skipped exactly).
__global__ __launch_bounds__(256)
void pack_lora_kernel(const float* __restrict__ lora_A, const float* __restrict__ lora_B,
                      const int* __restrict__ taskp,
                      float* __restrict__ A_pack, float* __restrict__ B_pack)
{
    const int idx  = blockIdx.x * 256 + threadIdx.x;
    const int task = taskp[0];
    int nvalid = (task >= 0) ? (task + 1) * 10 : 0;   // R = 10
    if (nvalid > 100) nvalid = 100;                   // T = 10

    if (idx < 128 * 1024) {                           // A_pack
        const int j = idx >> 10;
        A_pack[idx] = (j < nvalid) ? lora_A[idx] : 0.0f;  // lora_A[j*1024 + i]
    } else {                                          // B_pack
        const int k = idx - 128 * 1024;
        const int o = k >> 7;
        const int j = k & 127;
        float v = 0.0f;
        if (j < nvalid) {
            const int t = j / 10, r = j % 10;
            v = lora_B[((size_t)t * 1024 + o) * 10 + r];
        }
        B_pack[k] = v;
    }
}

extern "C" void kernel_launch(void* const* d_in, const int* in_sizes, int n_in,
                              void* d_out, int out_size, void* d_ws, size_t ws_size,
                              hipStream_t stream)
{
    const float* x      = (const float*)d_in[0];
    const float* U1_w   = (const float*)d_in[1];
    const float* U1_b   = (const float*)d_in[2];
    const float* U2_w   = (const float*)d_in[3];
    const float* U2_b   = (const float*)d_in[4];
    const float* U3_w   = (const float*)d_in[5];
    const float* U3_b   = (const float*)d_in[6];
    const float* C_w    = (const float*)d_in[7];
    const float* C_b    = (const float*)d_in[8];
    const float* ln1_g  = (const float*)d_in[9];
    const float* ln1_b  = (const float*)d_in[10];
    const float* ln3_g  = (const float*)d_in[11];
    const float* ln3_b  = (const float*)d_in[12];
    const float* alpha  = (const float*)d_in[13];
    const float* lora_A = (const float*)d_in[14];
    const float* lora_B = (const float*)d_in[15];
    const int*   taskp  = (const int*)d_in[16];
    float* out = (float*)d_out;

    const int Dh = in_sizes[2];           // 1024
    const int Dq = in_sizes[4];           // 128
    const int Do = in_sizes[8];           // 1024
    const int D  = in_sizes[1] / Dh;      // 1024
    const int NT = in_sizes[0] / D;       // 12544 tokens (= 98 * 128)
    const int JL = 128;                   // padded LoRA rank dim

    // Workspace layout (floats)
    float* out1   = (float*)d_ws;                 // [NT, Dh]  (becomes o1 in place)
    float* out3   = out1 + (size_t)NT * Dh;       // [NT, Dh]
    float* out2   = out3 + (size_t)NT * Dh;       // [NT, Dq]  (reused as z later)
    float* A_pack = out2 + (size_t)NT * Dq;       // [JL, D]
    float* B_pack = A_pack + (size_t)JL * D;      // [Do, JL]
    float* zbuf   = out2;                         // out2 dead after GEMM3

    dim3 blk(256);

    // 1) out1 = x . U1^T + b1
    gemm_f32_wmma<<<dim3(Dh / TN, NT / TM), blk, 0, stream>>>(x, U1_w, U1_b, out1, NT, Dh, D, 0);
    // 2) out2 = x . U2^T + b2
    gemm_f32_wmma<<<dim3(Dq / TN, NT / TM), blk, 0, stream>>>(x, U2_w, U2_b, out2, NT, Dq, D, 0);
    // 3) out3 = out2 . U3^T + b3
    gemm_f32_wmma<<<dim3(Dh / TN, NT / TM), blk, 0, stream>>>(out2, U3_w, U3_b, out3, NT, Dh, Dq, 0);
    // 4) o1 = LN(out1); o3 = LN(out3); o1 += alpha * o1 * o3   (in place over out1)
    ln_gate_kernel<<<dim3(NT / 8), blk, 0, stream>>>(out1, out3, ln1_g, ln1_b, ln3_g, ln3_b, alpha, NT, Dh);
    // 5) pack LoRA factors (reads task on-device; zeros if task < 0)
    pack_lora_kernel<<<dim3((JL * D + Do * JL) / 256), blk, 0, stream>>>(lora_A, lora_B, taskp, A_pack, B_pack);
    // 6) out = o1 . C^T + C_b
    gemm_f32_wmma<<<dim3(Do / TN, NT / TM), blk, 0, stream>>>(out1, C_w, C_b, out, NT, Do, Dh, 0);
    // 7) z = x . A_pack^T
    gemm_f32_wmma<<<dim3(JL / TN, NT / TM), blk, 0, stream>>>(x, A_pack, nullptr, zbuf, NT, JL, D, 0);
    // 8) out += z . B_pack^T
    gemm_f32_wmma<<<dim3(Do / TN, NT / TM), blk, 0, stream>>>(zbuf, B_pack, nullptr, out, NT, Do, JL, 1);
}